// GAT_747324309854
// MI455X (gfx1250) — compile-verified
//
#include <hip/hip_runtime.h>
#include <math.h>

#define N_NODES 50000
#define N_EDGES 800000
#define GDIM 128
#define HD1 128
#define HEADS 4
#define HD2 32
#define RREL 8
#define NB 20

typedef __attribute__((ext_vector_type(2))) float v2f;
typedef __attribute__((ext_vector_type(8))) float v8f;

__device__ __forceinline__ float lrelu(float x) { return x > 0.f ? x : 0.2f * x; }

// ordered-float atomic max: works for mixed signs given init = -inf
__device__ __forceinline__ void atomicMaxF(float* addr, float val) {
    if (val >= 0.f) atomicMax((int*)addr, __float_as_int(val));
    else            atomicMin((unsigned int*)addr, __float_as_uint(val));
}

// ---------------- small prep kernels ----------------

__global__ void k_basis_w(const float* __restrict__ comp, const float* __restrict__ basis,
                          float* __restrict__ W) {
    int t = blockIdx.x * blockDim.x + threadIdx.x;
    if (t >= RREL * GDIM * HD1) return;
    int r = t / (GDIM * HD1);
    int io = t % (GDIM * HD1);
    float s = 0.f;
    #pragma unroll
    for (int b = 0; b < NB; ++b) s += comp[r * NB + b] * basis[b * GDIM * HD1 + io];
    W[t] = s;
}

__global__ void k_count(const int* __restrict__ ei, const int* __restrict__ et,
                        float* __restrict__ cnt) {
    int e = blockIdx.x * blockDim.x + threadIdx.x;
    if (e >= N_EDGES) return;
    atomicAdd(&cnt[ei[N_EDGES + e] * RREL + et[e]], 1.f);
}

__global__ void k_fill(float* __restrict__ p, float v, int n) {
    int i = blockIdx.x * blockDim.x + threadIdx.x;
    if (i < n) p[i] = v;
}

__global__ void k_fill_bias(float* __restrict__ p, const float* __restrict__ bias, int n) {
    int i = blockIdx.x * blockDim.x + threadIdx.x;
    if (i < n) p[i] = bias[i & 127];
}

// ---------------- RGCN scatter (mean via pre-divided counts) ----------------

__global__ void k_scatter_rel(const float* __restrict__ x, const int* __restrict__ ei,
                              const int* __restrict__ et, const float* __restrict__ cnt,
                              float* __restrict__ acc, int r) {
    int gid = blockIdx.x * blockDim.x + threadIdx.x;
    int e = gid >> 5;
    if (e >= N_EDGES) return;
    if (et[e] != r) return;                  // uniform across the wave
    int lane = gid & 31;
    int s = ei[e], d = ei[N_EDGES + e];
    float scale = 1.f / cnt[d * RREL + r];
    float4 v = ((const float4*)(x + (size_t)s * GDIM))[lane];
    float* o = acc + (size_t)d * GDIM + lane * 4;
    atomicAdd(o + 0, v.x * scale);
    atomicAdd(o + 1, v.y * scale);
    atomicAdd(o + 2, v.z * scale);
    atomicAdd(o + 3, v.w * scale);
}

// ---------------- WMMA f32 matmul: C[M,128] (+)= A[M,128] @ B[128,128] ----------------
// BIAS_MODE=false: C initialized from existing C (accumulate)
// BIAS_MODE=true : C initialized from bias[col] (nullptr bias -> 0)

template <bool BIAS_MODE>
__global__ __launch_bounds__(256) void k_mm16(const float* __restrict__ A,
                                              const float* __restrict__ B,
                                              float* __restrict__ C,
                                              const float* __restrict__ bias) {
    __shared__ float As[16][132];            // stride 132: conflict-free A-frag reads
    int tx = threadIdx.x;                    // 0..31 (one wave along x)
    int ty = threadIdx.y;                    // 0..7  (8 waves -> 8 col tiles)
    int tid = ty * 32 + tx;
    int row0 = blockIdx.x * 16;

    const float4* A4 = (const float4*)(A + (size_t)row0 * 128);
    for (int i = tid; i < 16 * 32; i += 256) {
        int rr = i >> 5, cc = i & 31;
        float4 v = A4[rr * 32 + cc];
        As[rr][cc * 4 + 0] = v.x;
        As[rr][cc * 4 + 1] = v.y;
        As[rr][cc * 4 + 2] = v.z;
        As[rr][cc * 4 + 3] = v.w;
    }
    __syncthreads();

    int col0 = ty * 16;
    int half = tx >> 4;                      // lane group 0/1
    int mn = tx & 15;                        // A row / B,C col within tile
    int kg = half * 2;                       // K sub-offset for A/B frags

    v8f c;
    if (BIAS_MODE) {
        float bv = bias ? bias[col0 + mn] : 0.f;
        #pragma unroll
        for (int i = 0; i < 8; ++i) c[i] = bv;
    } else {
        #pragma unroll
        for (int i = 0; i < 8; ++i)
            c[i] = C[(size_t)(row0 + half * 8 + i) * 128 + col0 + mn];
    }

    #pragma unroll 4
    for (int k0 = 0; k0 < 128; k0 += 4) {
        v2f a, b;
        a.x = As[mn][k0 + kg];
        a.y = As[mn][k0 + kg + 1];
        b.x = B[(size_t)(k0 + kg) * 128 + col0 + mn];
        b.y = B[(size_t)(k0 + kg + 1) * 128 + col0 + mn];
        c = __builtin_amdgcn_wmma_f32_16x16x4_f32(false, a, false, b, (short)0, c,
                                                  false, false);
    }

    #pragma unroll
    for (int i = 0; i < 8; ++i)
        C[(size_t)(row0 + half * 8 + i) * 128 + col0 + mn] = c[i];
}

// ---------------- GAT attention ----------------

__global__ void k_attn(const float* __restrict__ xl, const float* __restrict__ att_src,
                       const float* __restrict__ att_dst, float* __restrict__ a_src,
                       float* __restrict__ a_dst) {
    int t = blockIdx.x * blockDim.x + threadIdx.x;
    if (t >= N_NODES * HEADS) return;
    int n = t >> 2, h = t & 3;
    const float* row = xl + (size_t)n * 128 + h * HD2;
    float ss = 0.f, sd = 0.f;
    #pragma unroll
    for (int j = 0; j < HD2; ++j) {
        float v = row[j];
        ss += v * att_src[h * HD2 + j];
        sd += v * att_dst[h * HD2 + j];
    }
    a_src[t] = ss;
    a_dst[t] = sd;
}

__global__ void k_edge_max(const int* __restrict__ ei, const float* __restrict__ a_src,
                           const float* __restrict__ a_dst, float* __restrict__ m) {
    int i = blockIdx.x * blockDim.x + threadIdx.x;
    if (i >= N_EDGES + N_NODES) return;
    int s, d;
    if (i < N_EDGES) { s = ei[i]; d = ei[N_EDGES + i]; if (s == d) return; }
    else             { s = d = i - N_EDGES; }
    #pragma unroll
    for (int h = 0; h < HEADS; ++h) {
        float v = lrelu(a_src[s * 4 + h] + a_dst[d * 4 + h]);
        atomicMaxF(&m[d * 4 + h], v);
    }
}

__global__ void k_edge_sum(const int* __restrict__ ei, const float* __restrict__ a_src,
                           const float* __restrict__ a_dst, const float* __restrict__ m,
                           float* __restrict__ denom) {
    int i = blockIdx.x * blockDim.x + threadIdx.x;
    if (i >= N_EDGES + N_NODES) return;
    int s, d;
    if (i < N_EDGES) { s = ei[i]; d = ei[N_EDGES + i]; if (s == d) return; }
    else             { s = d = i - N_EDGES; }
    #pragma unroll
    for (int h = 0; h < HEADS; ++h) {
        float v = lrelu(a_src[s * 4 + h] + a_dst[d * 4 + h]);
        atomicAdd(&denom[d * 4 + h], __expf(v - m[d * 4 + h]));
    }
}

__global__ void k_edge_scatter(const int* __restrict__ ei, const float* __restrict__ xl,
                               const float* __restrict__ a_src, const float* __restrict__ a_dst,
                               const float* __restrict__ m, const float* __restrict__ denom,
                               float* __restrict__ out) {
    int gid = blockIdx.x * blockDim.x + threadIdx.x;
    int i = gid >> 5;
    if (i >= N_EDGES + N_NODES) return;
    int lane = gid & 31;
    int s, d;
    if (i < N_EDGES) { s = ei[i]; d = ei[N_EDGES + i]; if (s == d) return; }
    else             { s = d = i - N_EDGES; }
    int h = lane >> 3;                       // 4 feats/lane -> 8 lanes per head
    float v = lrelu(a_src[s * 4 + h] + a_dst[d * 4 + h]);
    float w = __expf(v - m[d * 4 + h]) / denom[d * 4 + h];
    float4 xv = ((const float4*)(xl + (size_t)s * 128))[lane];
    float* o = out + (size_t)d * 128 + lane * 4;
    atomicAdd(o + 0, xv.x * w);
    atomicAdd(o + 1, xv.y * w);
    atomicAdd(o + 2, xv.z * w);
    atomicAdd(o + 3, xv.w * w);
}

// ---------------- host launch ----------------

extern "C" void kernel_launch(void* const* d_in, const int* in_sizes, int n_in,
                              void* d_out, int out_size, void* d_ws, size_t ws_size,
                              hipStream_t stream) {
    const float* x        = (const float*)d_in[0];
    const int*   ei       = (const int*)d_in[1];   // [0..E)=src, [E..2E)=dst
    const int*   et       = (const int*)d_in[2];
    const float* comp     = (const float*)d_in[3];
    const float* basis    = (const float*)d_in[4];
    const float* root     = (const float*)d_in[5];
    const float* bias1    = (const float*)d_in[6];
    const float* gat_w    = (const float*)d_in[7];
    const float* att_src  = (const float*)d_in[8];
    const float* att_dst  = (const float*)d_in[9];
    const float* gat_bias = (const float*)d_in[10];
    float* out = (float*)d_out;

    float* ws = (float*)d_ws;
    size_t off = 0;
    float* W    = ws + off; off += (size_t)RREL * GDIM * HD1;     // 131072
    float* cnt  = ws + off; off += (size_t)N_NODES * RREL;        // 400000
    float* hbuf = ws + off; off += (size_t)N_NODES * HD1;         // 6.4M
    float* xl   = ws + off; off += (size_t)N_NODES * HD1;         // 6.4M (also acc)
    float* asg  = ws + off; off += (size_t)N_NODES * HEADS;
    float* adg  = ws + off; off += (size_t)N_NODES * HEADS;
    float* mbuf = ws + off; off += (size_t)N_NODES * HEADS;
    float* dbuf = ws + off; off += (size_t)N_NODES * HEADS;
    float* acc = xl;

    dim3 mmBlock(32, 8);
    int mmGrid = N_NODES / 16;               // 3125, exact

    // --- RGCN ---
    k_basis_w<<<(RREL * GDIM * HD1 + 255) / 256, 256, 0, stream>>>(comp, basis, W);
    hipMemsetAsync(cnt, 0, (size_t)N_NODES * RREL * sizeof(float), stream);
    k_count<<<(N_EDGES + 255) / 256, 256, 0, stream>>>(ei, et, cnt);
    k_fill_bias<<<(N_NODES * HD1 + 255) / 256, 256, 0, stream>>>(hbuf, bias1, N_NODES * HD1);

    for (int r = 0; r < RREL; ++r) {
        hipMemsetAsync(acc, 0, (size_t)N_NODES * GDIM * sizeof(float), stream);
        k_scatter_rel<<<(N_EDGES * 32 + 255) / 256, 256, 0, stream>>>(x, ei, et, cnt, acc, r);
        k_mm16<false><<<mmGrid, mmBlock, 0, stream>>>(acc, W + (size_t)r * GDIM * HD1,
                                                      hbuf, nullptr);
    }
    k_mm16<false><<<mmGrid, mmBlock, 0, stream>>>(x, root, hbuf, nullptr);

    // --- GAT ---
    k_mm16<true><<<mmGrid, mmBlock, 0, stream>>>(hbuf, gat_w, xl, nullptr);  // xl = h @ W
    k_attn<<<(N_NODES * HEADS + 255) / 256, 256, 0, stream>>>(xl, att_src, att_dst, asg, adg);
    k_fill<<<(N_NODES * HEADS + 255) / 256, 256, 0, stream>>>(mbuf, -INFINITY, N_NODES * HEADS);
    hipMemsetAsync(dbuf, 0, (size_t)N_NODES * HEADS * sizeof(float), stream);

    int EN = N_EDGES + N_NODES;
    k_edge_max<<<(EN + 255) / 256, 256, 0, stream>>>(ei, asg, adg, mbuf);
    k_edge_sum<<<(EN + 255) / 256, 256, 0, stream>>>(ei, asg, adg, mbuf, dbuf);
    k_fill_bias<<<(N_NODES * HD1 + 255) / 256, 256, 0, stream>>>(out, gat_bias, N_NODES * HD1);
    k_edge_scatter<<<(EN * 32 + 255) / 256, 256, 0, stream>>>(ei, xl, asg, adg, mbuf, dbuf, out);

    (void)in_sizes; (void)n_in; (void)out_size; (void)ws_size;
}